// MacaqueBrain_56478819942642
// MI455X (gfx1250) — compile-verified
//
#include <hip/hip_runtime.h>

// ---------------------------------------------------------------------------
// Macaque-brain IF step:  ain = outs @ W ;  ain[160:192] += x ;
//   v_new = v + ain ; spikes = (v_new >= 1) ; v_out = v_new*(1-spikes)
//
// Roofline: W is 12256^2 fp32 = ~600MB, 2N^2 = 0.3 GFLOP -> 0.5 FLOP/byte,
// pure HBM-bound (23.3 TB/s -> ~26us floor). outs is binary (~50% ones), so
// skipping inactive source rows halves W traffic (~300MB, ~13us floor).
// Strategy: (1) deterministic compaction of active source indices,
// (2) gathered row-sum GEMV via V_WMMA_F32_16X16X4_F32 (A row0 = ones),
// split-K over 8 waves/block + LDS reduce + fused IF epilogue.
// ---------------------------------------------------------------------------

#define NN      12256          // total neurons
#define IN_LO   160            // IN_AREA*NSZ
#define IN_HI   192
#define SLICES  8              // split-K waves per block
#define CGROUPS 383            // NN / 32 column-groups (32 cols per block)

typedef float v2f __attribute__((ext_vector_type(2)));
typedef float v8f __attribute__((ext_vector_type(8)));

// ---------------------------------------------------------------------------
// Kernel 1: deterministic (sorted) compaction of indices where outs[i] != 0.
// One block of 256 threads; each thread owns a contiguous slab of 48 rows,
// counts, LDS exclusive scan, then writes its indices at its offset.
// ---------------------------------------------------------------------------
__global__ __launch_bounds__(256)
void snn_compact_kernel(const float* __restrict__ outs,
                        int* __restrict__ idx,
                        int* __restrict__ count) {
    __shared__ int s_cnt[256];
    const int t   = threadIdx.x;
    const int per = (NN + 255) / 256;         // 48
    const int b0  = t * per;
    const int b1  = (b0 + per < NN) ? (b0 + per) : NN;

    int c = 0;
    for (int i = b0; i < b1; ++i) c += (outs[i] != 0.0f) ? 1 : 0;
    s_cnt[t] = c;
    __syncthreads();

    if (t == 0) {                              // serial exclusive scan (256 elems, trivial)
        int run = 0;
        for (int i = 0; i < 256; ++i) { int v = s_cnt[i]; s_cnt[i] = run; run += v; }
        count[0] = run;
    }
    __syncthreads();

    int off = s_cnt[t];
    for (int i = b0; i < b1; ++i)
        if (outs[i] != 0.0f) idx[off++] = i;
}

// ---------------------------------------------------------------------------
// Kernel 2: row-gather GEMV via WMMA + fused IF update.
// grid = 383 blocks (32 output columns each), block = (32,8) = 8 waves.
// Wave w sums its split-K slice of active rows into two 16x16 f32 WMMA
// accumulators (row 0 = result). A-matrix row 0 = 1.0 in all four k-slots
// (lanes 0 and 16, both VGPRs), so the WMMA computes sum of 4 gathered rows
// regardless of the exact k<->(vgpr,half) striping of the B operand.
// ---------------------------------------------------------------------------
__global__ __launch_bounds__(256)
void snn_wmma_kernel(const float* __restrict__ W,
                     const float* __restrict__ x,
                     const float* __restrict__ v,
                     const int*   __restrict__ idx,
                     const int*   __restrict__ count,
                     float*       __restrict__ out) {
    const int lane    = threadIdx.x;                                   // 0..31
    const int wv      = __builtin_amdgcn_readfirstlane((int)threadIdx.y); // 0..7 (SGPR)
    const int colbase = blockIdx.x * 32;
    const int nlo     = lane & 15;
    const int half    = lane >> 4;

    // Active-row count (SGPR-uniform); slice rounded to multiple of 4 so the
    // int4 index loads stay 16B-aligned and chunks are full inside the loop.
    const int m   = __builtin_amdgcn_readfirstlane(count[0]);
    int per       = (m + SLICES - 1) / SLICES;
    per           = (per + 3) & ~3;
    int k0        = wv * per;       if (k0 > m) k0 = m;
    int k1        = k0 + per;       if (k1 > m) k1 = m;

    v8f acc0 = {0.f,0.f,0.f,0.f,0.f,0.f,0.f,0.f};
    v8f acc1 = {0.f,0.f,0.f,0.f,0.f,0.f,0.f,0.f};
    v2f amain;                      // A row 0 = 1.0 (lanes 0 & 16), rest 0
    amain.x = (nlo == 0) ? 1.0f : 0.0f;
    amain.y = amain.x;

    const float* W0 = W + colbase + nlo;       // tile0 column for this lane
    const float* W1 = W0 + 16;                 // tile1 column

    int j = k0;
#pragma unroll 2
    for (; j + 4 <= k1; j += 4) {
        const int4 iv = *(const int4*)(idx + j);   // 16B-aligned, wave-uniform
        const size_t ra = (size_t)(half ? iv.z : iv.x) * (size_t)NN;
        const size_t rb = (size_t)(half ? iv.w : iv.y) * (size_t)NN;
        v2f b0; b0.x = W0[ra]; b0.y = W0[rb];
        v2f b1; b1.x = W1[ra]; b1.y = W1[rb];
        acc0 = __builtin_amdgcn_wmma_f32_16x16x4_f32(
                   false, amain, false, b0, (short)0, acc0, false, false);
        acc1 = __builtin_amdgcn_wmma_f32_16x16x4_f32(
                   false, amain, false, b1, (short)0, acc1, false, false);
    }

    // Tail (<=3 rows, wave-uniform count): plain VALU adds (layout-agnostic).
    float e0 = 0.0f, e1 = 0.0f;
    const int rem = k1 - j;
    for (int t = 0; t < rem; ++t) {
        const size_t r = (size_t)idx[j + t] * (size_t)NN;
        e0 += W0[r];
        e1 += W1[r];
    }

    // C/D layout: VGPR0 lanes 0..15 hold M=0, N=lane  -> our result row.
    __shared__ float s_red[SLICES][32];
    if (lane < 16) {
        s_red[wv][lane]      = acc0[0] + e0;
        s_red[wv][16 + lane] = acc1[0] + e1;
    }
    __syncthreads();

    if (wv == 0) {                 // SGPR-uniform branch; 32 lanes, 32 columns
        float s = 0.0f;
        for (int w2 = 0; w2 < SLICES; ++w2) s += s_red[w2][lane];
        const int c = colbase + lane;
        if (c >= IN_LO && c < IN_HI) s += x[c - IN_LO];
        const float vn = v[c] + s;
        const float sp = (vn >= 1.0f) ? 1.0f : 0.0f;
        out[c]      = sp;                 // spikes
        out[NN + c] = vn * (1.0f - sp);   // v_out (hard reset on spike)
    }
}

// ---------------------------------------------------------------------------
// d_in order (setup_inputs): x[32], outs[N], v[N], W[N*N]  (all fp32)
// d_out: spikes[N] ++ v_out[N]
// d_ws layout: [0..3] count (int), [16 .. 16+4*NN) index list
// ---------------------------------------------------------------------------
extern "C" void kernel_launch(void* const* d_in, const int* in_sizes, int n_in,
                              void* d_out, int out_size, void* d_ws, size_t ws_size,
                              hipStream_t stream) {
    const float* x    = (const float*)d_in[0];
    const float* outs = (const float*)d_in[1];
    const float* v    = (const float*)d_in[2];
    const float* W    = (const float*)d_in[3];
    float* out        = (float*)d_out;

    int* count = (int*)d_ws;
    int* idx   = (int*)((char*)d_ws + 16);

    snn_compact_kernel<<<1, 256, 0, stream>>>(outs, idx, count);

    dim3 blk(32, SLICES);
    snn_wmma_kernel<<<CGROUPS, blk, 0, stream>>>(W, x, v, idx, count, out);
}